// CRSM_987842478111
// MI455X (gfx1250) — compile-verified
//
#include <hip/hip_runtime.h>
#include <hip/hip_bf16.h>

typedef __attribute__((ext_vector_type(2))) float v2f;
typedef __attribute__((ext_vector_type(8))) float v8f;
typedef __attribute__((ext_vector_type(4))) int   v4i;

#define N_NODES 100000
#define N_EDGES 1600000

// LDS pitches chosen so hi-half-wave rows (+2 rows / +8 rows) map to a bank
// window disjoint from the lo-half-wave's 16-wide window (2*pitch % 64 == 32,
// 8*20 % 64 == 32).
#define W1_PITCH 144
#define W2_PITCH 48
#define HS_PITCH 20

// ---------------------------------------------------------------- zero scratch
__global__ void crsm_zero(float* __restrict__ p, int n) {
    int i = blockIdx.x * blockDim.x + threadIdx.x;
    if (i < n) p[i] = 0.0f;
}

// ------------------------------------------------------------- edge scatter
// One thread per undirected edge; does both directions.
// conical[node] = x[node][16:32] -> one 64B cacheline, gathered via 4x b128.
// ~54M f32 atomics total, all hitting a 6.8MB region resident in the 192MB L2.
__global__ void crsm_scatter(const float* __restrict__ x,
                             const int* __restrict__ ei,
                             float* __restrict__ ns,
                             float* __restrict__ deg) {
    int e = blockIdx.x * blockDim.x + threadIdx.x;
    if (e >= N_EDGES) return;
    int i = ei[e];
    int j = ei[N_EDGES + e];

    const float4* xi = (const float4*)(x + (size_t)i * 32 + 16);
    const float4* xj = (const float4*)(x + (size_t)j * 32 + 16);
    float4 ci[4], cj[4];
#pragma unroll
    for (int q = 0; q < 4; ++q) { ci[q] = xi[q]; cj[q] = xj[q]; }

    float* nsi = ns + (size_t)i * 16;
    float* nsj = ns + (size_t)j * 16;
#pragma unroll
    for (int q = 0; q < 4; ++q) {
        __hip_atomic_fetch_add(nsi + 4*q + 0, cj[q].x, __ATOMIC_RELAXED, __HIP_MEMORY_SCOPE_AGENT);
        __hip_atomic_fetch_add(nsi + 4*q + 1, cj[q].y, __ATOMIC_RELAXED, __HIP_MEMORY_SCOPE_AGENT);
        __hip_atomic_fetch_add(nsi + 4*q + 2, cj[q].z, __ATOMIC_RELAXED, __HIP_MEMORY_SCOPE_AGENT);
        __hip_atomic_fetch_add(nsi + 4*q + 3, cj[q].w, __ATOMIC_RELAXED, __HIP_MEMORY_SCOPE_AGENT);
        __hip_atomic_fetch_add(nsj + 4*q + 0, ci[q].x, __ATOMIC_RELAXED, __HIP_MEMORY_SCOPE_AGENT);
        __hip_atomic_fetch_add(nsj + 4*q + 1, ci[q].y, __ATOMIC_RELAXED, __HIP_MEMORY_SCOPE_AGENT);
        __hip_atomic_fetch_add(nsj + 4*q + 2, ci[q].z, __ATOMIC_RELAXED, __HIP_MEMORY_SCOPE_AGENT);
        __hip_atomic_fetch_add(nsj + 4*q + 3, ci[q].w, __ATOMIC_RELAXED, __HIP_MEMORY_SCOPE_AGENT);
    }
    __hip_atomic_fetch_add(deg + i, 1.0f, __ATOMIC_RELAXED, __HIP_MEMORY_SCOPE_AGENT);
    __hip_atomic_fetch_add(deg + j, 1.0f, __ATOMIC_RELAXED, __HIP_MEMORY_SCOPE_AGENT);
}

// ------------------------------------------------------------------ fused MLP
// Block = 256 threads = 8 waves; each wave owns a 16-node tile.
// GEMM1: combined(16x32) @ W1(32x128), relu, then GEMM2 accumulated per
// 16-wide h tile: out(16x32) += h_t(16x16) @ W2[16t:16t+16][:].
// All matrix math via V_WMMA_F32_16X16X4_F32 (exact f32).
__launch_bounds__(256)
__global__ void crsm_mlp(const float* __restrict__ x,
                         const float* __restrict__ ns,
                         const float* __restrict__ deg,
                         const float* __restrict__ W1,
                         const float* __restrict__ b1,
                         const float* __restrict__ W2,
                         const float* __restrict__ b2,
                         float* __restrict__ out) {
    __shared__ float w1s[32 * W1_PITCH];
    __shared__ float w2s[128 * W2_PITCH];
    __shared__ float hs[8][16 * HS_PITCH];

    // ---- Stage weights into LDS once per block.
#if __has_builtin(__builtin_amdgcn_global_load_async_to_lds_b128)
    // CDNA5 async copy path: GLOBAL_LOAD_ASYNC_TO_LDS_B128 (ASYNCcnt-tracked,
    // no VGPR round-trip). Rows (128 / 32 floats) are float4-divisible, so a
    // B128 chunk never straddles a pitched LDS row.
    for (int t = threadIdx.x; t < (32 * 128) / 4; t += 256) {
        int idx = t * 4;
        int r = idx >> 7, c = idx & 127;
        __builtin_amdgcn_global_load_async_to_lds_b128(
            (__attribute__((address_space(1))) v4i*)(W1 + idx),
            (__attribute__((address_space(3))) v4i*)&w1s[r * W1_PITCH + c],
            0, 0);
    }
    for (int t = threadIdx.x; t < (128 * 32) / 4; t += 256) {
        int idx = t * 4;
        int r = idx >> 5, c = idx & 31;
        __builtin_amdgcn_global_load_async_to_lds_b128(
            (__attribute__((address_space(1))) v4i*)(W2 + idx),
            (__attribute__((address_space(3))) v4i*)&w2s[r * W2_PITCH + c],
            0, 0);
    }
#if __has_builtin(__builtin_amdgcn_s_wait_asynccnt)
    __builtin_amdgcn_s_wait_asynccnt(0);
#else
    asm volatile("s_wait_asynccnt 0x0" ::: "memory");
#endif
#else
    for (int t = threadIdx.x; t < 32 * 128; t += 256) {
        int r = t >> 7, c = t & 127;
        w1s[r * W1_PITCH + c] = W1[t];
    }
    for (int t = threadIdx.x; t < 128 * 32; t += 256) {
        int r = t >> 5, c = t & 31;
        w2s[r * W2_PITCH + c] = W2[t];
    }
#endif
    __syncthreads();

    const int wave = threadIdx.x >> 5;
    const int lane = threadIdx.x & 31;
    const int lo   = lane & 15;
    const int hi   = lane >> 4;          // 0 = lanes 0-15, 1 = lanes 16-31
    const int row0 = (blockIdx.x * 8 + wave) * 16;
    const int node = row0 + lo;
    const bool valid = node < N_NODES;

    // --- A fragments for GEMM1 (16x4 f32 A layout: lo half K={0,1}, hi half K={2,3})
    float invdeg = 0.0f;
    if (valid) invdeg = 1.0f / fmaxf(deg[node], 1.0f);
    v2f a1[8];
#pragma unroll
    for (int s = 0; s < 8; ++s) {
        int c = 4 * s + 2 * hi;
        float vx = 0.0f, vy = 0.0f;
        if (valid) {
            if (c < 16) {                 // radial = x[:, 0:16]
                vx = x[(size_t)node * 32 + c];
                vy = x[(size_t)node * 32 + c + 1];
            } else {                      // agg = ns / clip(deg,1)
                vx = ns[(size_t)node * 16 + (c - 16)] * invdeg;
                vy = ns[(size_t)node * 16 + (c - 15)] * invdeg;
            }
        }
        a1[s].x = vx; a1[s].y = vy;
    }

    // Output accumulators (16x32 = two 16x16 C tiles), initialized with b2.
    v8f acc_out[2];
#pragma unroll
    for (int t2 = 0; t2 < 2; ++t2) {
        float bb = b2[16 * t2 + lo];
#pragma unroll
        for (int v = 0; v < 8; ++v) acc_out[t2][v] = bb;
    }

    float* myhs = hs[wave];

#pragma unroll
    for (int t = 0; t < 8; ++t) {
        // ---- GEMM1 N-tile t: h = relu(A1 @ W1[:, 16t:16t+16] + b1)
        v8f acc;
        {
            float bb = b1[16 * t + lo];
#pragma unroll
            for (int v = 0; v < 8; ++v) acc[v] = bb;
        }
#pragma unroll
        for (int s = 0; s < 8; ++s) {
            int rbase = 4 * s + 2 * hi;   // B layout: v0 = K rows {0 lo, 2 hi}, v1 = {1 lo, 3 hi}
            v2f b;
            b.x = w1s[(rbase + 0) * W1_PITCH + 16 * t + lo];
            b.y = w1s[(rbase + 1) * W1_PITCH + 16 * t + lo];
            acc = __builtin_amdgcn_wmma_f32_16x16x4_f32(
                false, a1[s], false, b, (short)0, acc, false, false);
        }
#pragma unroll
        for (int v = 0; v < 8; ++v) acc[v] = fmaxf(acc[v], 0.0f);

        // ---- C-layout -> A-layout via per-wave LDS tile.
        // hs is private to this wave and same-wave LDS ops execute in order
        // (DScnt, in-order per ISA), so a compiler scheduling fence is enough —
        // no block-wide barrier needed.
        __builtin_amdgcn_wave_barrier();
#pragma unroll
        for (int v = 0; v < 8; ++v)
            myhs[(v + 8 * hi) * HS_PITCH + lo] = acc[v];
        __builtin_amdgcn_wave_barrier();

        // ---- GEMM2: out(16x32) += h_t(16x16) @ W2[16t:16t+16][0:32]
#pragma unroll
        for (int s2 = 0; s2 < 4; ++s2) {
            int c = 4 * s2 + 2 * hi;
            v2f a2;
            a2.x = myhs[lo * HS_PITCH + c];
            a2.y = myhs[lo * HS_PITCH + c + 1];
            int rbase = 16 * t + 4 * s2 + 2 * hi;
#pragma unroll
            for (int t2 = 0; t2 < 2; ++t2) {
                v2f b;
                b.x = w2s[(rbase + 0) * W2_PITCH + 16 * t2 + lo];
                b.y = w2s[(rbase + 1) * W2_PITCH + 16 * t2 + lo];
                acc_out[t2] = __builtin_amdgcn_wmma_f32_16x16x4_f32(
                    false, a2, false, b, (short)0, acc_out[t2], false, false);
            }
        }
        __builtin_amdgcn_wave_barrier();
    }

    // ---- store out (C layout: M = v + 8*hi, N = 16*t2 + lo) — coalesced b32s
#pragma unroll
    for (int t2 = 0; t2 < 2; ++t2) {
#pragma unroll
        for (int v = 0; v < 8; ++v) {
            int r = row0 + v + 8 * hi;
            if (r < N_NODES)
                out[(size_t)r * 32 + 16 * t2 + lo] = acc_out[t2][v];
        }
    }
}

// ------------------------------------------------------------------- launcher
extern "C" void kernel_launch(void* const* d_in, const int* in_sizes, int n_in,
                              void* d_out, int out_size, void* d_ws, size_t ws_size,
                              hipStream_t stream) {
    const float* x  = (const float*)d_in[0];
    const int*   ei = (const int*)  d_in[1];
    const float* W1 = (const float*)d_in[2];
    const float* b1 = (const float*)d_in[3];
    const float* W2 = (const float*)d_in[4];
    const float* b2 = (const float*)d_in[5];
    float* out = (float*)d_out;

    float* ns  = (float*)d_ws;                       // N_NODES * 16
    float* deg = ns + (size_t)N_NODES * 16;          // N_NODES

    int zn = N_NODES * 17;
    crsm_zero<<<(zn + 255) / 256, 256, 0, stream>>>((float*)d_ws, zn);
    crsm_scatter<<<(N_EDGES + 255) / 256, 256, 0, stream>>>(x, ei, ns, deg);
    crsm_mlp<<<(N_NODES + 127) / 128, 256, 0, stream>>>(x, ns, deg, W1, b1, W2, b2, out);
}